// GatedGraphCrossAttentionLayer_54065048322663
// MI455X (gfx1250) — compile-verified
//
#include <hip/hip_runtime.h>
#include <cstdint>
#include <cstddef>

typedef __attribute__((ext_vector_type(2))) float v2f;
typedef __attribute__((ext_vector_type(8))) float v8f;
typedef __attribute__((ext_vector_type(4))) int   v4i;

#define B_    16
#define S_    4096
#define D_    512
#define NEL   2048
#define NEDGE 8192

// Async global->LDS 16-byte copy (CDNA5: GLOBAL_LOAD_ASYNC_TO_LDS_B128, ASYNCcnt)
__device__ __forceinline__ void async_cp16(const float* g, float* l) {
    v4i* gg = (v4i*)const_cast<float*>(g);   // pointee reinterpret (generic)
    v4i* ll = (v4i*)l;
    __builtin_amdgcn_global_load_async_to_lds_b128(
        (__attribute__((address_space(1))) v4i*)gg,   // addrspacecast -> global
        (__attribute__((address_space(3))) v4i*)ll,   // addrspacecast -> LDS
        0, 0);
}

// ---------------------------------------------------------------------------
// Degree / normalization kernels
// ---------------------------------------------------------------------------
__global__ void zero_deg_kernel(int* __restrict__ deg) {
    int i = blockIdx.x * blockDim.x + threadIdx.x;
    if (i < B_ * NEL) deg[i] = 0;
}

__global__ void count_deg_kernel(const int* __restrict__ ei, int* __restrict__ deg) {
    int i = blockIdx.x * blockDim.x + threadIdx.x;
    if (i >= B_ * NEDGE) return;
    int b = i / NEDGE;
    int e = i - b * NEDGE;
    int dst = ei[(size_t)b * 2 * NEDGE + NEDGE + e];
    atomicAdd(&deg[b * NEL + dst], 1);
}

__global__ void make_dis_kernel(const int* __restrict__ deg, float* __restrict__ dis) {
    int i = blockIdx.x * blockDim.x + threadIdx.x;
    if (i < B_ * NEL) dis[i] = rsqrtf((float)deg[i] + 1.0f);  // +1 = self loop
}

// ---------------------------------------------------------------------------
// Fused gather + GEMM (x = gather(t) @ W^T) with fp32 WMMA.
// Double-buffered async global->LDS staging overlaps the next k-block's
// copies with the current k-block's WMMAs. Epilogue writes x and initializes
// out = x * dis^2 + bias (self-loop contribution).
// Block = 128 threads (4 waves); each block computes a 64x64 tile.
// ---------------------------------------------------------------------------
__global__ __launch_bounds__(128)
void gemm_gather_wmma_kernel(const float* __restrict__ t, const float* __restrict__ W,
                             const float* __restrict__ bias, const int* __restrict__ t2e,
                             const float* __restrict__ dis,
                             float* __restrict__ x, float* __restrict__ out)
{
    const int bm   = blockIdx.x;          // row block  (NEL/64 = 32)
    const int bn   = blockIdx.y;          // col block  (D/64   = 8)
    const int b    = blockIdx.z;          // batch
    const int tid  = threadIdx.x;         // 0..127
    const int wave = tid >> 5;            // 0..3 -> M sub-tile
    const int lane = tid & 31;
    const int l15  = lane & 15;
    const int kh   = (lane >> 4) << 1;    // lanes 0-15: K=0,1 ; lanes 16-31: K=2,3

    constexpr int LSTR = 36;              // padded LDS row stride (floats)
    __shared__ float As[2][64 * LSTR];    // gathered A rows [m][k], double-buffered
    __shared__ float Ws[2][64 * LSTR];    // W rows (= B cols) [n][k]

    // Per-thread staging assignment: 4 x 16B chunks of A and of W per k-block.
    const float* arow[4];
    const float* wrow[4];
    int ldsoff[4];
    const float* Wbase = W + (size_t)(bn * 64) * D_;
#pragma unroll
    for (int it = 0; it < 4; ++it) {
        int f4 = tid + it * 128;          // 0..511
        int m  = f4 >> 3;                 // 8 float4 per 32-float row
        int c  = (f4 & 7) << 2;
        int grow = t2e[b * NEL + bm * 64 + m];           // gather fused into GEMM
        arow[it]   = t + ((size_t)b * S_ + grow) * D_ + c;
        wrow[it]   = Wbase + (size_t)m * D_ + c;
        ldsoff[it] = m * LSTR + c;
    }

    v8f acc[4];
#pragma unroll
    for (int i = 0; i < 4; ++i) acc[i] = v8f{0, 0, 0, 0, 0, 0, 0, 0};

    // Prologue: async-stage k-block 0 into buffer 0 (8 copies per thread)
#pragma unroll
    for (int it = 0; it < 4; ++it) {
        async_cp16(arow[it], &As[0][ldsoff[it]]);
        async_cp16(wrow[it], &Ws[0][ldsoff[it]]);
    }

    for (int kb = 0; kb < 16; ++kb) {     // 16 k-blocks of 32
        const int cur = kb & 1;
        if (kb + 1 < 16) {
            const int nb = (kb + 1) & 1;
            const int k0 = (kb + 1) * 32;
#pragma unroll
            for (int it = 0; it < 4; ++it) {
                async_cp16(arow[it] + k0, &As[nb][ldsoff[it]]);
                async_cp16(wrow[it] + k0, &Ws[nb][ldsoff[it]]);
            }
            // Wait until only the 8 just-issued copies remain -> current buffer done
            asm volatile("s_wait_asynccnt 0x8" ::: "memory");
        } else {
            asm volatile("s_wait_asynccnt 0x0" ::: "memory");
        }
        __syncthreads();                  // make all waves' copies visible

        const float* Ac = As[cur];
        const float* Wc = Ws[cur];
#pragma unroll
        for (int kk = 0; kk < 32; kk += 4) {
            // A fragment: 16x4 fp32 (ISA layout: VGPR0=K0/K2, VGPR1=K1/K3)
            v2f a = *(const v2f*)&Ac[(wave * 16 + l15) * LSTR + kk + kh];
#pragma unroll
            for (int nt = 0; nt < 4; ++nt) {
                // B fragment: 4x16 fp32, B[k][n] = W[n][k]
                v2f bf = *(const v2f*)&Wc[(nt * 16 + l15) * LSTR + kk + kh];
                acc[nt] = __builtin_amdgcn_wmma_f32_16x16x4_f32(
                    false, a, false, bf, (short)0, acc[nt], false, false);
            }
        }
        __syncthreads();                  // buffer may be overwritten next iter
    }

    // Epilogue: C/D layout -> row = r + 8*(lane>=16), col = lane&15
    const int mrow0 = bm * 64 + wave * 16 + ((lane >> 4) << 3);
#pragma unroll
    for (int nt = 0; nt < 4; ++nt) {
        const int col = bn * 64 + nt * 16 + l15;
        const float bv = bias[col];
#pragma unroll
        for (int r = 0; r < 8; ++r) {
            const int row = mrow0 + r;
            const float d1 = dis[b * NEL + row];
            const size_t off = ((size_t)b * NEL + row) * D_ + col;
            const float v = acc[nt][r];
            x[off]   = v;
            out[off] = v * d1 * d1 + bv;  // self-loop norm = dis^2, plus bias
        }
    }
}

// ---------------------------------------------------------------------------
// Edge message passing: out[dst] += x[src] * dis[src]*dis[dst]
// One block per edge, fp32 global atomics across D.
// ---------------------------------------------------------------------------
__global__ __launch_bounds__(256)
void edge_scatter_kernel(const int* __restrict__ ei, const float* __restrict__ dis,
                         const float* __restrict__ x, float* __restrict__ out)
{
    int blk = blockIdx.x;                 // b*NEDGE + e
    int b = blk / NEDGE;
    int e = blk - b * NEDGE;
    int src = ei[(size_t)b * 2 * NEDGE + e];
    int dst = ei[(size_t)b * 2 * NEDGE + NEDGE + e];
    float w = dis[b * NEL + src] * dis[b * NEL + dst];
    const float* xs = x   + ((size_t)b * NEL + src) * D_;
    float*       od = out + ((size_t)b * NEL + dst) * D_;
    for (int d = threadIdx.x; d < D_; d += 256)
        atomicAdd(&od[d], xs[d] * w);
}

// ---------------------------------------------------------------------------
// y = t  (float4 copy), then scatter y[b, e2t[j]] = t + tanh(gate) * out[b, j]
// ---------------------------------------------------------------------------
__global__ void copy_t_kernel(const float4* __restrict__ src, float4* __restrict__ dst, int n4) {
    int i = blockIdx.x * blockDim.x + threadIdx.x;
    if (i < n4) dst[i] = src[i];
}

__global__ __launch_bounds__(128)
void scatter_final_kernel(const float* __restrict__ t, const float* __restrict__ gate,
                          const int* __restrict__ e2t, const float* __restrict__ out,
                          float* __restrict__ y)
{
    int blk = blockIdx.x;                 // b*NEL + j
    int b = blk / NEL;
    int j = blk - b * NEL;
    float g = tanhf(gate[0]);
    int s = e2t[b * NEL + j];
    const float4* tv = (const float4*)(t   + ((size_t)b * S_  + s) * D_);
    const float4* ov = (const float4*)(out + ((size_t)b * NEL + j) * D_);
    float4*       yv = (float4*)      (y   + ((size_t)b * S_  + s) * D_);
    for (int d = threadIdx.x; d < D_ / 4; d += 128) {
        float4 a = tv[d], o = ov[d];
        yv[d] = float4{a.x + g * o.x, a.y + g * o.y, a.z + g * o.z, a.w + g * o.w};
    }
}

// ---------------------------------------------------------------------------
extern "C" void kernel_launch(void* const* d_in, const int* in_sizes, int n_in,
                              void* d_out, int out_size, void* d_ws, size_t ws_size,
                              hipStream_t stream) {
    const float* t    = (const float*)d_in[0];   // [B,S,D]
    const float* W    = (const float*)d_in[1];   // [D,D]
    const float* bias = (const float*)d_in[2];   // [D]
    const float* gate = (const float*)d_in[3];   // [1]
    const int*   t2e  = (const int*)d_in[4];     // [B,NEL]
    const int*   e2t  = (const int*)d_in[5];     // [B,NEL]
    const int*   ei   = (const int*)d_in[6];     // [B,2,NEDGE]
    float* y = (float*)d_out;

    // Workspace layout
    char* ws = (char*)d_ws;
    const size_t XB = (size_t)B_ * NEL * D_ * sizeof(float);   // 64 MB
    float* x   = (float*)(ws);
    float* out = (float*)(ws + XB);
    int*   deg = (int*)  (ws + 2 * XB);
    float* dis = (float*)(ws + 2 * XB + (size_t)B_ * NEL * sizeof(int));

    // 1-3: degree -> dis = rsqrt(deg + 1)
    zero_deg_kernel<<<(B_ * NEL + 255) / 256, 256, 0, stream>>>(deg);
    count_deg_kernel<<<(B_ * NEDGE + 255) / 256, 256, 0, stream>>>(ei, deg);
    make_dis_kernel<<<(B_ * NEL + 255) / 256, 256, 0, stream>>>(deg, dis);

    // 4: fused gather + fp32 WMMA GEMM + (self-loop + bias) epilogue
    dim3 gg(NEL / 64, D_ / 64, B_);
    gemm_gather_wmma_kernel<<<gg, 128, 0, stream>>>(t, W, bias, t2e, dis, x, out);

    // 5: edge aggregation via atomics
    edge_scatter_kernel<<<B_ * NEDGE, 256, 0, stream>>>(ei, dis, x, out);

    // 6-7: residual copy + gated scatter-back
    int n4 = B_ * S_ * D_ / 4;
    copy_t_kernel<<<(n4 + 255) / 256, 256, 0, stream>>>((const float4*)t, (float4*)y, n4);
    scatter_final_kernel<<<B_ * NEL, 128, 0, stream>>>(t, gate, e2t, out, y);
}